// unKG_GSL_85890755985724
// MI455X (gfx1250) — compile-verified
//
#include <hip/hip_runtime.h>
#include <hip/hip_bf16.h>

typedef __attribute__((ext_vector_type(2))) float v2f;
typedef __attribute__((ext_vector_type(8))) float v8f;

#define N_NODES 100000
#define N_REL   8
#define N_EDGES 100000
#define DIM     128
#define TILES   (N_EDGES / 16)   // 6250, exact
#define AS_STRIDE 132            // 128 + 4 pad -> conflict-free ds_load_b64

// ---- B-matrix preload: lane holds B[k+koff+v][col] pairs per the ISA
//      4x16 f32 B layout (lanes 0-15: K=k..k+1, lanes 16-31: K=k+2..k+3).
__device__ __forceinline__ void load_B_regs(const float* __restrict__ W,
                                            int col, int koff, v2f* bW) {
#pragma unroll
  for (int kk = 0; kk < 32; ++kk) {
    int k = 4 * kk + koff;
    v2f b;
    b.x = W[(size_t)k * DIM + col];
    b.y = W[(size_t)(k + 1) * DIM + col];
    bW[kk] = b;
  }
}

// ---- 16x16 f32 tile: 32 chained V_WMMA_F32_16X16X4_F32 over K=128.
__device__ __forceinline__ v8f mma_tile_f32(const float* __restrict__ As,
                                            int ln, int koff, const v2f* bW) {
  v8f c = {0.f, 0.f, 0.f, 0.f, 0.f, 0.f, 0.f, 0.f};
#pragma unroll
  for (int kk = 0; kk < 32; ++kk) {
    // A 16x4 f32 layout: lane m (=ln) reads K = 4*kk + koff, +1
    v2f a = *(const v2f*)&As[ln * AS_STRIDE + 4 * kk + koff];
    c = __builtin_amdgcn_wmma_f32_16x16x4_f32(false, a, false, bW[kk],
                                              (short)0, c, false, false);
  }
  return c;
}

// ================= dense self-transform: out = x @ W (N x 128 @ 128 x 128)
__global__ __launch_bounds__(256) void self_gemm_kernel(
    const float* __restrict__ x, const float* __restrict__ W,
    float* __restrict__ out) {
  __shared__ float As[16 * AS_STRIDE];
  const int tid  = threadIdx.x;
  const int lane = tid & 31;
  const int wave = tid >> 5;           // 8 waves -> 8 column tiles
  const int ln   = lane & 15;
  const int hi   = lane >> 4;
  const int koff = hi * 2;
  const int n0   = wave * 16;

  v2f bW[32];
  load_B_regs(W, n0 + ln, koff, bW);

  for (int t = blockIdx.x; t < TILES; t += gridDim.x) {
    // cooperative A-tile load: 16 rows x 128 cols = 512 float4
#pragma unroll
    for (int j = tid; j < 512; j += 256) {
      int row = j >> 5, c4 = j & 31;
      float4 v = *(const float4*)(x + (size_t)(t * 16 + row) * DIM + c4 * 4);
      *(float4*)&As[row * AS_STRIDE + c4 * 4] = v;
    }
    __syncthreads();

    v8f c = mma_tile_f32(As, ln, koff, bW);

#pragma unroll
    for (int v = 0; v < 8; ++v) {
      int row = v + 8 * hi;            // C/D layout: VGPR v, halves 8 apart
      out[(size_t)(t * 16 + row) * DIM + n0 + ln] = c[v];
    }
    __syncthreads();
  }
}

// ================= edge messages: gather src rows, GEMM with relW[r],
//                   scale by edge weight, atomic scatter-add to out[tgt].
__global__ __launch_bounds__(256) void edge_gemm_scatter_kernel(
    const float* __restrict__ x, const int* __restrict__ edge_index,
    const float* __restrict__ ew, const float* __restrict__ relW,
    float* __restrict__ out) {
  __shared__ float As[16 * AS_STRIDE];
  __shared__ int   sTgt[16];
  __shared__ float sWt[16];

  const int r = blockIdx.y;
  const float* __restrict__ W    = relW + (size_t)r * DIM * DIM;
  const int*   __restrict__ srcp = edge_index + (size_t)(2 * r) * N_EDGES;
  const int*   __restrict__ tgtp = srcp + N_EDGES;
  const float* __restrict__ wp   = ew + (size_t)r * N_EDGES;

  const int tid  = threadIdx.x;
  const int lane = tid & 31;
  const int wave = tid >> 5;
  const int ln   = lane & 15;
  const int hi   = lane >> 4;
  const int koff = hi * 2;
  const int n0   = wave * 16;

  v2f bW[32];
  load_B_regs(W, n0 + ln, koff, bW);

  for (int t = blockIdx.x; t < TILES; t += gridDim.x) {
    if (tid < 16) {
      int e = t * 16 + tid;
      sTgt[tid] = tgtp[e];
      sWt[tid]  = wp[e];
    }
    // gathered A-tile: row = edge-in-tile, redundant src-index reads hit L1
#pragma unroll
    for (int j = tid; j < 512; j += 256) {
      int row = j >> 5, c4 = j & 31;
      int s = srcp[t * 16 + row];
      float4 v = *(const float4*)(x + (size_t)s * DIM + c4 * 4);
      *(float4*)&As[row * AS_STRIDE + c4 * 4] = v;
    }
    __syncthreads();

    v8f c = mma_tile_f32(As, ln, koff, bW);

#pragma unroll
    for (int v = 0; v < 8; ++v) {
      int row = v + 8 * hi;
      float val = c[v] * sWt[row];
      atomicAdd(&out[(size_t)sTgt[row] * DIM + n0 + ln], val);
    }
    __syncthreads();
  }
}

// ================= per-(relation,node) target flags for the bias mask
__global__ void zero_words_kernel(int* __restrict__ p, int n) {
  int gid = blockIdx.x * blockDim.x + threadIdx.x;
  if (gid < n) p[gid] = 0;
}

__global__ void set_flags_kernel(const int* __restrict__ edge_index,
                                 unsigned char* __restrict__ flags) {
  int gid = blockIdx.x * blockDim.x + threadIdx.x;
  if (gid < N_REL * N_EDGES) {
    int r = gid / N_EDGES, e = gid - r * N_EDGES;
    int tgt = edge_index[(size_t)(2 * r + 1) * N_EDGES + e];
    flags[(size_t)r * N_NODES + tgt] = 1;  // races benign: all write 1
  }
}

// ================= bias add via mask + optional ReLU (in place)
__global__ void bias_act_kernel(float* __restrict__ out,
                                const float* __restrict__ bias,
                                const unsigned char* __restrict__ flags,
                                int do_relu) {
  int gid = blockIdx.x * blockDim.x + threadIdx.x;   // one float4 per thread
  if (gid >= N_NODES * (DIM / 4)) return;
  int n = gid >> 5, c4 = gid & 31;
  float4 v = ((const float4*)out)[gid];
#pragma unroll
  for (int r = 0; r < N_REL; ++r) {
    if (flags[(size_t)r * N_NODES + n]) {
      float4 b = ((const float4*)bias)[r * (DIM / 4) + c4];
      v.x += b.x; v.y += b.y; v.z += b.z; v.w += b.w;
    }
  }
  if (do_relu) {
    v.x = fmaxf(v.x, 0.f); v.y = fmaxf(v.y, 0.f);
    v.z = fmaxf(v.z, 0.f); v.w = fmaxf(v.w, 0.f);
  }
  ((float4*)out)[gid] = v;
}

extern "C" void kernel_launch(void* const* d_in, const int* in_sizes, int n_in,
                              void* d_out, int out_size, void* d_ws, size_t ws_size,
                              hipStream_t stream) {
  const float* x      = (const float*)d_in[0];
  const int*   eidx   = (const int*)d_in[1];
  const float* ew     = (const float*)d_in[2];
  const float* relW1  = (const float*)d_in[3];
  const float* selfW1 = (const float*)d_in[4];
  const float* bias1  = (const float*)d_in[5];
  const float* relW2  = (const float*)d_in[6];
  const float* selfW2 = (const float*)d_in[7];
  const float* bias2  = (const float*)d_in[8];
  float* out = (float*)d_out;

  float* h1 = (float*)d_ws;                                   // 51.2 MB
  unsigned char* flags =
      (unsigned char*)d_ws + (size_t)N_NODES * DIM * sizeof(float);  // 0.8 MB

  const int GEMM_BLOCKS = 512;       // persistent: ~12 tiles/block
  const int flag_words = (N_REL * N_NODES + 3) / 4;

  // ---- target flags (identical for both layers)
  zero_words_kernel<<<(flag_words + 255) / 256, 256, 0, stream>>>((int*)flags,
                                                                  flag_words);
  set_flags_kernel<<<(N_REL * N_EDGES + 255) / 256, 256, 0, stream>>>(eidx,
                                                                      flags);

  // ---- layer 1: h1 = relu(x@selfW1 + scatter(msgs) + mask*bias1)
  self_gemm_kernel<<<GEMM_BLOCKS, 256, 0, stream>>>(x, selfW1, h1);
  edge_gemm_scatter_kernel<<<dim3(GEMM_BLOCKS, N_REL), 256, 0, stream>>>(
      x, eidx, ew, relW1, h1);
  bias_act_kernel<<<(N_NODES * (DIM / 4) + 255) / 256, 256, 0, stream>>>(
      h1, bias1, flags, 1);

  // ---- layer 2: out = h1@selfW2 + scatter(msgs) + mask*bias2
  self_gemm_kernel<<<GEMM_BLOCKS, 256, 0, stream>>>(h1, selfW2, out);
  edge_gemm_scatter_kernel<<<dim3(GEMM_BLOCKS, N_REL), 256, 0, stream>>>(
      h1, eidx, ew, relW2, out);
  bias_act_kernel<<<(N_NODES * (DIM / 4) + 255) / 256, 256, 0, stream>>>(
      out, bias2, flags, 0);
}